// CrispComposition_23106924052651
// MI455X (gfx1250) — compile-verified
//
#include <hip/hip_runtime.h>
#include <stdint.h>

// CrispComposition: out[b,i] = max_o min(m[b,i], w[i,o]) == min(m[b,i], max_o w[i,o])
// B=4096, IN=512, OUT=256. Memory-bound: ~16.5 MB traffic -> ~0.7us @ 23.3 TB/s.

#define IN_DIM 512
#define OUT_DIM 256
#define VEC_PER_ROW (IN_DIM / 4)   // 128 float4 per m-row
#define ITER 4                     // float4 elements per thread in crisp_min

// ---------------------------------------------------------------------------
// Kernel B (hot): out[b,i] = min(m[b,i], W[i]).
// Stage W (2 KB) into LDS via CDNA5 async global->LDS DMA (ASYNCcnt path).
// Since stride % VEC_PER_ROW == 0 (blockDim=256), each thread's W element is
// loop-invariant: one ds_load_b128 per thread, then a pure
// load_b128 / 4x v_min_num_f32 / store_b128 stream.
// ---------------------------------------------------------------------------
__global__ __launch_bounds__(256) void crisp_min(const float* __restrict__ m,
                                                 const float* __restrict__ wmax,
                                                 float* __restrict__ out,
                                                 int nvec4) {
    __shared__ float sW[IN_DIM];

    // 128 lanes each async-copy 16B of W into LDS.
    // global_load_async_to_lds_b128 vdst(LDS byte addr), vaddr(64b addr), off
    if (threadIdx.x < VEC_PER_ROW) {
        uint32_t lds_off = (uint32_t)(uintptr_t)(&sW[threadIdx.x * 4]);
        uint64_t gaddr   = (uint64_t)(uintptr_t)(wmax + threadIdx.x * 4);
        asm volatile("global_load_async_to_lds_b128 %0, %1, off"
                     :
                     : "v"(lds_off), "v"(gaddr)
                     : "memory");
    }
    asm volatile("s_wait_asynccnt 0x0" ::: "memory");
    __syncthreads();

    const int tid0   = blockIdx.x * blockDim.x + threadIdx.x;
    const int stride = gridDim.x * blockDim.x;          // multiple of 128

    // Loop-invariant W fragment for this thread (row phase of tid0).
    const float4 wv = ((const float4*)sW)[tid0 & (VEC_PER_ROW - 1)];

    const float4* __restrict__ m4 = (const float4*)m;
    float4* __restrict__ o4       = (float4*)out;

    int    idx[ITER];
    float4 mv[ITER];
#pragma unroll
    for (int k = 0; k < ITER; ++k)
        idx[k] = tid0 + k * stride;

    // Issue all loads first -> one s_clause of independent global_load_b128.
#pragma unroll
    for (int k = 0; k < ITER; ++k)
        if (idx[k] < nvec4) mv[k] = m4[idx[k]];

#pragma unroll
    for (int k = 0; k < ITER; ++k)
        if (idx[k] < nvec4) {
            float4 r;
            r.x = fminf(mv[k].x, wv.x);
            r.y = fminf(mv[k].y, wv.y);
            r.z = fminf(mv[k].z, wv.z);
            r.w = fminf(mv[k].w, wv.w);
            o4[idx[k]] = r;
        }
}

// ---------------------------------------------------------------------------
// Kernel A (one-shot, 0.5 MB): W[i] = max over 256 cols of weight[i, :].
// One wave32 per row; lane reduces 8 floats, then 5-step ds_swizzle XOR tree.
// ds_swizzle group-of-32 encoding: and=0x1F | or<<5 | xor<<10.
// ---------------------------------------------------------------------------
__device__ __forceinline__ float wave32_max(float x) {
    x = fmaxf(x, __int_as_float(__builtin_amdgcn_ds_swizzle(__float_as_int(x), 0x401F))); // xor 16
    x = fmaxf(x, __int_as_float(__builtin_amdgcn_ds_swizzle(__float_as_int(x), 0x201F))); // xor 8
    x = fmaxf(x, __int_as_float(__builtin_amdgcn_ds_swizzle(__float_as_int(x), 0x101F))); // xor 4
    x = fmaxf(x, __int_as_float(__builtin_amdgcn_ds_swizzle(__float_as_int(x), 0x081F))); // xor 2
    x = fmaxf(x, __int_as_float(__builtin_amdgcn_ds_swizzle(__float_as_int(x), 0x041F))); // xor 1
    return x;
}

__global__ __launch_bounds__(256) void crisp_rowmax(const float* __restrict__ w,
                                                    float* __restrict__ wmax) {
    const int lane = threadIdx.x & 31;
    const int wave = threadIdx.x >> 5;
    const int row  = (blockIdx.x << 3) + wave;          // < 512
    const float4* wr = (const float4*)(w + (size_t)row * OUT_DIM);
    float4 a = wr[lane * 2 + 0];
    float4 b = wr[lane * 2 + 1];
    float mx = fmaxf(fmaxf(fmaxf(a.x, a.y), fmaxf(a.z, a.w)),
                     fmaxf(fmaxf(b.x, b.y), fmaxf(b.z, b.w)));
    mx = wave32_max(mx);
    if (lane == 0) wmax[row] = mx;
}

extern "C" void kernel_launch(void* const* d_in, const int* in_sizes, int n_in,
                              void* d_out, int out_size, void* d_ws, size_t ws_size,
                              hipStream_t stream) {
    const float* m = (const float*)d_in[0];   // [B, IN] fp32
    const float* w = (const float*)d_in[1];   // [IN, OUT] fp32
    float* out  = (float*)d_out;              // [B, IN] fp32
    float* wmax = (float*)d_ws;               // [IN] scratch row-max

    // Kernel A: 512 rows, 8 rows (8 waves) per block.
    crisp_rowmax<<<IN_DIM / 8, 256, 0, stream>>>(w, wmax);

    // Kernel B: nvec4 float4 elements, ITER per thread.
    int nvec4  = out_size / 4;                            // 524288
    int blocks = (nvec4 + 256 * ITER - 1) / (256 * ITER); // 512
    crisp_min<<<blocks, 256, 0, stream>>>(m, wmax, out, nvec4);
}